// WJS_55319178772781
// MI455X (gfx1250) — compile-verified
//
#include <hip/hip_runtime.h>
#include <stdint.h>

#define COLS    4096
#define THREADS 256
#define NWAVES  (THREADS / 32)
#define F4ROW   (COLS / 4)         // 1024 float4 per row
#define F4THR   (F4ROW / THREADS)  // 4 float4 per thread per matrix

// ---------------- wave32 reductions ----------------
__device__ __forceinline__ float wave_sum(float v) {
#pragma unroll
  for (int m = 16; m >= 1; m >>= 1) v += __shfl_xor(v, m, 32);
  return v;
}
__device__ __forceinline__ float wave_max(float v) {
#pragma unroll
  for (int m = 16; m >= 1; m >>= 1) v = fmaxf(v, __shfl_xor(v, m, 32));
  return v;
}

// Block-wide reduction of two values (returns result to all threads).
template <bool IS_MAX>
__device__ __forceinline__ void block_red2(float& x, float& y,
                                           volatile float* sA, volatile float* sB,
                                           int tid) {
  x = IS_MAX ? wave_max(x) : wave_sum(x);
  y = IS_MAX ? wave_max(y) : wave_sum(y);
  const int wave = tid >> 5, lane = tid & 31;
  if (lane == 0) { sA[wave] = x; sB[wave] = y; }
  __syncthreads();
  if (tid == 0) {
    float ax = sA[0], ay = sB[0];
#pragma unroll
    for (int w = 1; w < NWAVES; ++w) {
      ax = IS_MAX ? fmaxf(ax, sA[w]) : ax + sA[w];
      ay = IS_MAX ? fmaxf(ay, sB[w]) : ay + sB[w];
    }
    sA[0] = ax; sB[0] = ay;
  }
  __syncthreads();
  x = sA[0]; y = sB[0];
  __syncthreads();  // allow sA/sB reuse by the next pass
}

// ---------------- per-row kernel ----------------
// One block per row: async-stage both 16KB rows to LDS (GVS mode, shared
// IOFFSET advances LDS dst and global src together), then register-resident
// three-pass softmax/JSD reduction. Writes sqrt(row inner sum) to rowOut[row].
__global__ __launch_bounds__(THREADS)
void jsd_row_kernel(const float* __restrict__ P, const float* __restrict__ Q,
                    float* __restrict__ rowOut) {
  __shared__ float4 ldsP[F4ROW];
  __shared__ float4 ldsQ[F4ROW];
  __shared__ float  sA[NWAVES], sB[NWAVES];

  const int tid = threadIdx.x;
  const int row = blockIdx.x;
  const float* pRow = P + (size_t)row * COLS;
  const float* qRow = Q + (size_t)row * COLS;

  // ---- CDNA5 async global -> LDS staging (ASYNCcnt path) ----
  // GVS addressing: saddr = row base (SGPR pair), vaddr = tid*16 byte offset.
  // IOFFSET is added to BOTH the LDS address (VDST vgpr) and the global
  // address, so one LDS-base VGPR + offset immediates cover all 4 chunks.
  {
    const uint32_t lp0  = (uint32_t)(uintptr_t)(&ldsP[tid]);  // low 32b == LDS offset
    const uint32_t lq0  = (uint32_t)(uintptr_t)(&ldsQ[tid]);
    const uint32_t voff = (uint32_t)tid * 16u;
    asm volatile(
        "global_load_async_to_lds_b128 %0, %2, %3 offset:0 th:TH_LOAD_NT\n\t"
        "global_load_async_to_lds_b128 %0, %2, %3 offset:4096 th:TH_LOAD_NT\n\t"
        "global_load_async_to_lds_b128 %0, %2, %3 offset:8192 th:TH_LOAD_NT\n\t"
        "global_load_async_to_lds_b128 %0, %2, %3 offset:12288 th:TH_LOAD_NT\n\t"
        "global_load_async_to_lds_b128 %1, %2, %4 offset:0 th:TH_LOAD_NT\n\t"
        "global_load_async_to_lds_b128 %1, %2, %4 offset:4096 th:TH_LOAD_NT\n\t"
        "global_load_async_to_lds_b128 %1, %2, %4 offset:8192 th:TH_LOAD_NT\n\t"
        "global_load_async_to_lds_b128 %1, %2, %4 offset:12288 th:TH_LOAD_NT\n\t"
        "s_wait_asynccnt 0"
        :
        : "v"(lp0), "v"(lq0), "v"(voff), "s"(pRow), "s"(qRow)
        : "memory");
  }
  // Each thread reads back only chunks its own wave issued; s_wait_asynccnt 0
  // already ordered those, so no block barrier is needed here.

  // ---- LDS -> registers: 16 + 16 elements per thread ----
  float4 a4[F4THR], b4[F4THR], d4[F4THR];
#pragma unroll
  for (int g = 0; g < F4THR; ++g) {
    const int idx = tid + g * THREADS;
    a4[g] = ldsP[idx];
    b4[g] = ldsQ[idx];
  }

  // ---- pass 1: row maxes, cache (p-q)^2 ----
  float maxp = -__builtin_huge_valf(), maxq = -__builtin_huge_valf();
#pragma unroll
  for (int g = 0; g < F4THR; ++g) {
    const float* a = (const float*)&a4[g];
    const float* b = (const float*)&b4[g];
    float*       d = (float*)&d4[g];
#pragma unroll
    for (int c = 0; c < 4; ++c) {
      maxp = fmaxf(maxp, a[c]);
      maxq = fmaxf(maxq, b[c]);
      const float df = a[c] - b[c];
      d[c] = df * df;
    }
  }
  block_red2<true>(maxp, maxq, sA, sB, tid);

  // ---- pass 2: softmax denominators ----
  float sp = 0.f, sq = 0.f;
#pragma unroll
  for (int g = 0; g < F4THR; ++g) {
    const float* a = (const float*)&a4[g];
    const float* b = (const float*)&b4[g];
#pragma unroll
    for (int c = 0; c < 4; ++c) {
      sp += __expf(a[c] - maxp);
      sq += __expf(b[c] - maxq);
    }
  }
  block_red2<false>(sp, sq, sA, sB, tid);

  const float invsp = 1.0f / sp;
  const float invsq = 1.0f / sq;
  const float lsp   = __logf(sp);
  const float lsq   = __logf(sq);

  // ---- pass 3: JSD accumulate: log p = (a-max) - log Sp (only log m is trans) ----
  float acc = 0.f;
#pragma unroll
  for (int g = 0; g < F4THR; ++g) {
    const float* a = (const float*)&a4[g];
    const float* b = (const float*)&b4[g];
    const float* d = (const float*)&d4[g];
#pragma unroll
    for (int c = 0; c < 4; ++c) {
      const float ap = a[c] - maxp;
      const float bq = b[c] - maxq;
      const float p  = __expf(ap) * invsp;
      const float q  = __expf(bq) * invsq;
      const float m  = 0.5f * (p + q);
      const float lm = __logf(m);
      const float lp = ap - lsp;
      const float lq = bq - lsq;
      const float jsd = 0.5f * (q * (lq - lm) + p * (lp - lm));
      acc += d[c] * jsd;
    }
  }
  float dummy = 0.f;
  block_red2<false>(acc, dummy, sA, sB, tid);

  if (tid == 0) rowOut[row] = sqrtf(acc);
}

// ---------------- final deterministic sum over rows ----------------
__global__ __launch_bounds__(THREADS)
void final_sum_kernel(const float* __restrict__ rows, int nrows,
                      float* __restrict__ out) {
  __shared__ float sA[NWAVES], sB[NWAVES];
  const int tid = threadIdx.x;
  float acc = 0.f;
  for (int i = tid; i < nrows; i += THREADS) acc += rows[i];
  float dummy = 0.f;
  block_red2<false>(acc, dummy, sA, sB, tid);
  if (tid == 0) out[0] = acc;
}

// ---------------- host entry ----------------
extern "C" void kernel_launch(void* const* d_in, const int* in_sizes, int n_in,
                              void* d_out, int out_size, void* d_ws, size_t ws_size,
                              hipStream_t stream) {
  (void)n_in; (void)out_size; (void)ws_size;
  const float* P = (const float*)d_in[0];
  const float* Q = (const float*)d_in[1];
  const int rows = in_sizes[0] / COLS;  // 8192

  float* rowOut = (float*)d_ws;  // rows floats of scratch

  jsd_row_kernel<<<dim3(rows), dim3(THREADS), 0, stream>>>(P, Q, rowOut);
  final_sum_kernel<<<dim3(1), dim3(THREADS), 0, stream>>>(rowOut, rows, (float*)d_out);
}